// TGNMemory_63496796504586
// MI455X (gfx1250) — compile-verified
//
#include <hip/hip_runtime.h>
#include <cmath>

typedef __attribute__((ext_vector_type(2))) float v2f;
typedef __attribute__((ext_vector_type(8))) float v8f;

#define MEMD 128
#define XP 516          // LDS pitch for X (512 cols + 4 pad -> conflict-free A-fragment reads)
#define TILE 128        // nodes per workgroup (128*516*4 B = 258 KB LDS < 320 KB/WGP)

// ---------------- Kernel 1: zero the selection keys ----------------
__global__ void init_keys_kernel(unsigned long long* keys, int n) {
  int i = blockIdx.x * blockDim.x + threadIdx.x;
  if (i < n) keys[i] = 0ull;
}

// ---------------- Kernel 2: LastAggregator via 64-bit atomicMax ----------------
// key = (bits(t) << 32) | (idx+1): t >= 0 so float bits are order-preserving;
// ties in t resolved toward larger idx (matches stable argsort + segment_max).
__global__ void select_last_kernel(const int* __restrict__ dst,
                                   const float* __restrict__ t,
                                   unsigned long long* __restrict__ keys, int m) {
  int i = blockIdx.x * blockDim.x + threadIdx.x;
  if (i < m) {
    unsigned tb = __float_as_uint(t[i]);
    unsigned long long key = ((unsigned long long)tb << 32) | (unsigned)(i + 1);
    atomicMax(&keys[dst[i]], key);
  }
}

__device__ __forceinline__ v8f wmma4(v2f a, v2f b, v8f c) {
  return __builtin_amdgcn_wmma_f32_16x16x4_f32(false, a, false, b, (short)0, c,
                                               false, false);
}

// ---------------- Kernel 3: message build + GRU via f32 WMMA ----------------
// Wave w owns node-subtile w (16 nodes) and loops over 8 j-tiles; per j-tile it
// keeps 4 accumulators (r, z, i_n, h_n) so the GRU combine is done in-register.
__global__ __launch_bounds__(256, 1)
void gru_wmma_kernel(const float* __restrict__ memory,
                     const float* __restrict__ last_update,
                     const int* __restrict__ src,
                     const float* __restrict__ t,
                     const float* __restrict__ raw_msg,
                     const float* __restrict__ time_w,
                     const float* __restrict__ time_b,
                     const float* __restrict__ W_ih,
                     const float* __restrict__ W_hh,
                     const float* __restrict__ b_ih,
                     const float* __restrict__ b_hh,
                     const unsigned long long* __restrict__ keys,
                     float* __restrict__ out) {
  extern __shared__ float smem[];
  float* xs = smem;                         // [TILE][XP]: 0..127 mem[src] | 128..255 h=mem[node] | 256..383 raw | 384..511 t_enc
  int* hasmsg = (int*)(smem + TILE * XP);   // [TILE]

  const int tid = threadIdx.x;
  const int base = blockIdx.x * TILE;

  // ---- Phase 1: stage X into LDS (128 threads per row, 2 rows in flight) ----
  {
    const int c = tid & 127;
    for (int rr = tid >> 7; rr < TILE; rr += 2) {
      const int node = base + rr;
      unsigned long long key = keys[node];
      const int has = (key != 0ull);
      float* xr = xs + (size_t)rr * XP;
      xr[128 + c] = memory[(size_t)node * MEMD + c];   // h (also msg's mem[dst] part)
      if (has) {
        const int sel = (int)(unsigned)(key & 0xffffffffull) - 1;
        const int si = src[sel];
        xr[c]       = memory[(size_t)si * MEMD + c];
        xr[256 + c] = raw_msg[(size_t)sel * MEMD + c];
        const float rel = t[sel] - last_update[si];
        xr[384 + c] = cosf(rel * time_w[c] + time_b[c]);
      } else {
        xr[c] = 0.f; xr[256 + c] = 0.f; xr[384 + c] = 0.f;
      }
      if (c == 0) hasmsg[rr] = has;
    }
  }
  __syncthreads();

  // ---- Phase 2: WMMA GEMMs + in-register GRU combine ----
  {
    const int wave = tid >> 5, lane = tid & 31;
    const int half = lane >> 4, lm = lane & 15;
    const int off = 2 * half;
    const int msub = wave;                              // 0..7: 16-node subtile
    const float* xrowA = xs + (size_t)(msub * 16 + lm) * XP;  // A row base (msg, K=512)
    const float* xrowH = xrowA + 128;                         // A row base (h, K=128)

    for (int jt = 0; jt < 8; ++jt) {
      const int j0 = jt << 4;
      // B = W^T slices: lane holds column n = n_base + lm
      const float* wr = W_ih + (size_t)(j0 + lm) * 512;        // r gate, gi part
      const float* wz = W_ih + (size_t)(128 + j0 + lm) * 512;  // z gate, gi part
      const float* wi = W_ih + (size_t)(256 + j0 + lm) * 512;  // i_n
      const float* vr = W_hh + (size_t)(j0 + lm) * 128;        // r gate, gh part
      const float* vz = W_hh + (size_t)(128 + j0 + lm) * 128;  // z gate, gh part
      const float* vn = W_hh + (size_t)(256 + j0 + lm) * 128;  // h_n

      v8f ar = {0.f,0.f,0.f,0.f,0.f,0.f,0.f,0.f};
      v8f az = {0.f,0.f,0.f,0.f,0.f,0.f,0.f,0.f};
      v8f ai = {0.f,0.f,0.f,0.f,0.f,0.f,0.f,0.f};
      v8f ah = {0.f,0.f,0.f,0.f,0.f,0.f,0.f,0.f};

#pragma unroll 4
      for (int k = 0; k < 512; k += 4) {                // gi: K = 512 (msg)
        v2f a  = *(const v2f*)(xrowA + k + off);        // shared by 3 WMMAs
        v2f br = *(const v2f*)(wr + k + off);
        v2f bz = *(const v2f*)(wz + k + off);
        v2f bi = *(const v2f*)(wi + k + off);
        ar = wmma4(a, br, ar);
        az = wmma4(a, bz, az);
        ai = wmma4(a, bi, ai);
      }
#pragma unroll 4
      for (int k = 0; k < 128; k += 4) {                // gh: K = 128 (h)
        v2f a  = *(const v2f*)(xrowH + k + off);
        v2f br = *(const v2f*)(vr + k + off);
        v2f bz = *(const v2f*)(vz + k + off);
        v2f bh = *(const v2f*)(vn + k + off);
        ar = wmma4(a, br, ar);
        az = wmma4(a, bz, az);
        ah = wmma4(a, bh, ah);
      }

      // ---- GRU gates in registers; D layout: lane holds D[v + 8*half][lm] ----
      const int j = j0 + lm;
      const float br_ = b_ih[j] + b_hh[j];
      const float bz_ = b_ih[128 + j] + b_hh[128 + j];
      const float bi_ = b_ih[256 + j];
      const float bh_ = b_hh[256 + j];
#pragma unroll
      for (int v = 0; v < 8; ++v) {
        const int m = msub * 16 + v + 8 * half;
        const float h = xs[(size_t)m * XP + 128 + j];
        const int has = hasmsg[m];
        const float sr = ar[v] + br_;
        const float sz = az[v] + bz_;
        const float in_ = ai[v] + bi_;
        const float hn  = ah[v] + bh_;
        const float rg = 1.f / (1.f + expf(-sr));
        const float zg = 1.f / (1.f + expf(-sz));
        const float ng = tanhf(in_ + rg * hn);
        const float outv = has ? (1.f - zg) * ng + zg * h : h;
        out[(size_t)(base + m) * MEMD + j] = outv;
      }
    }
  }
}

extern "C" void kernel_launch(void* const* d_in, const int* in_sizes, int n_in,
                              void* d_out, int out_size, void* d_ws, size_t ws_size,
                              hipStream_t stream) {
  const float* memory      = (const float*)d_in[0];
  const float* last_update = (const float*)d_in[1];
  const int*   src         = (const int*)d_in[2];
  const int*   dst         = (const int*)d_in[3];
  const float* t           = (const float*)d_in[4];
  const float* raw_msg     = (const float*)d_in[5];
  const float* time_w      = (const float*)d_in[6];
  const float* time_b      = (const float*)d_in[7];
  const float* W_ih        = (const float*)d_in[8];
  const float* W_hh        = (const float*)d_in[9];
  const float* b_ih        = (const float*)d_in[10];
  const float* b_hh        = (const float*)d_in[11];
  float* out = (float*)d_out;

  const int num_nodes = in_sizes[0] / MEMD;
  const int n_msgs    = in_sizes[4];
  unsigned long long* keys = (unsigned long long*)d_ws;

  init_keys_kernel<<<(num_nodes + 255) / 256, 256, 0, stream>>>(keys, num_nodes);
  select_last_kernel<<<(n_msgs + 255) / 256, 256, 0, stream>>>(dst, t, keys, n_msgs);

  const int smem_bytes = TILE * XP * (int)sizeof(float) + TILE * (int)sizeof(int); // ~259 KB
  (void)hipFuncSetAttribute((const void*)gru_wmma_kernel,
                            hipFuncAttributeMaxDynamicSharedMemorySize, smem_bytes);
  gru_wmma_kernel<<<num_nodes / TILE, 256, smem_bytes, stream>>>(
      memory, last_update, src, t, raw_msg, time_w, time_b,
      W_ih, W_hh, b_ih, b_hh, keys, out);
}